// Object_sdf_81741817578227
// MI455X (gfx1250) — compile-verified
//
#include <hip/hip_runtime.h>
#include <math.h>

// MI455X / gfx1250, wave32. Brute-force signed-distance-to-mesh.
// Dot-product stage = V_WMMA_F32_16X16X4_F32; K slots = (x, y, z, pp),
// so the third WMMA against (-2a, 1) yields |p|^2 - 2 a.p in one shot.
// Per-triangle data = one 128B AoS record -> 5x global_load_b128 per tile.
// Best candidate tracked as packed u64 key: (f32bits(d2)<<32)|triIdx,
// so min() == argmin with first-index tie-breaking.

typedef float v2f __attribute__((ext_vector_type(2)));
typedef float v4f __attribute__((ext_vector_type(4)));
typedef float v8f __attribute__((ext_vector_type(8)));
typedef unsigned long long u64;
typedef unsigned int u32;

#define WAVES 32
#define TMAIN 1024

__device__ __forceinline__ v4f ld4(const float* base, u32 byteOff) {
    return *(const v4f*)((const char*)base + byteOff);
}
__device__ __forceinline__ v2f ld2(const float* base, u32 byteOff) {
    return *(const v2f*)((const char*)base + byteOff);
}

// ---------------- Pass 1: bbox -> center + 1/scale into ws[0..3] ----------------
__global__ void k_bbox(const float* __restrict__ verts, int V, float* __restrict__ ws) {
    __shared__ float sm[6][256];
    int t = threadIdx.x;
    float mnx = 1e30f, mny = 1e30f, mnz = 1e30f;
    float mxx = -1e30f, mxy = -1e30f, mxz = -1e30f;
    for (int i = t; i < V; i += 256) {
        float x = verts[3 * i + 0], y = verts[3 * i + 1], z = verts[3 * i + 2];
        mnx = fminf(mnx, x); mny = fminf(mny, y); mnz = fminf(mnz, z);
        mxx = fmaxf(mxx, x); mxy = fmaxf(mxy, y); mxz = fmaxf(mxz, z);
    }
    sm[0][t] = mnx; sm[1][t] = mny; sm[2][t] = mnz;
    sm[3][t] = mxx; sm[4][t] = mxy; sm[5][t] = mxz;
    __syncthreads();
    for (int s = 128; s > 0; s >>= 1) {
        if (t < s) {
            sm[0][t] = fminf(sm[0][t], sm[0][t + s]);
            sm[1][t] = fminf(sm[1][t], sm[1][t + s]);
            sm[2][t] = fminf(sm[2][t], sm[2][t + s]);
            sm[3][t] = fmaxf(sm[3][t], sm[3][t + s]);
            sm[4][t] = fmaxf(sm[4][t], sm[4][t + s]);
            sm[5][t] = fmaxf(sm[5][t], sm[5][t + s]);
        }
        __syncthreads();
    }
    if (t == 0) {
        float cx = 0.5f * (sm[0][0] + sm[3][0]);
        float cy = 0.5f * (sm[1][0] + sm[4][0]);
        float cz = 0.5f * (sm[2][0] + sm[5][0]);
        float sc = 0.5f * fmaxf(fmaxf(sm[3][0] - sm[0][0], sm[4][0] - sm[1][0]),
                                sm[5][0] - sm[2][0]);
        ws[0] = cx; ws[1] = cy; ws[2] = cz; ws[3] = 1.0f / sc;
    }
}

// ---------------- Pass 2: normalize queries into SoA qx/qy/qz + |p|^2 ----------------
__global__ void k_queries(const float* __restrict__ q, int N, float* __restrict__ ws) {
    int i = blockIdx.x * blockDim.x + threadIdx.x;
    if (i >= N) return;
    float cx = ws[0], cy = ws[1], cz = ws[2], inv = ws[3];
    float x = (q[3 * i + 0] - cx) * inv;
    float y = (q[3 * i + 1] - cy) * inv;
    float z = (q[3 * i + 2] - cz) * inv;
    float* qx = ws + 4;
    qx[i] = x; qx[N + i] = y; qx[2 * N + i] = z;
    qx[3 * N + i] = x * x + y * y + z * z;
}

// ---------------- Pass 3: per-triangle 128B AoS record ----------------
// rec[0..3]  = abx aby abz acx     rec[4..7]  = acy acz ax ay
// rec[8..11] = az  ab.a ac.a ab.b  rec[12..15]= ac.b ab.c ac.c |ab|^2
// rec[16..19]= |ac|^2 ab.ac |a|^2 nx          rec[20..21] = ny nz
__global__ void k_tris(const float* __restrict__ verts, const int* __restrict__ faces,
                       int F, int N, float* __restrict__ ws) {
    int f = blockIdx.x * blockDim.x + threadIdx.x;
    if (f >= F) return;
    float cx = ws[0], cy = ws[1], cz = ws[2], inv = ws[3];
    int ia = faces[3 * f + 0], ib = faces[3 * f + 1], ic = faces[3 * f + 2];
    float ax = (verts[3 * ia + 0] - cx) * inv, ay = (verts[3 * ia + 1] - cy) * inv, az = (verts[3 * ia + 2] - cz) * inv;
    float bx = (verts[3 * ib + 0] - cx) * inv, by = (verts[3 * ib + 1] - cy) * inv, bz = (verts[3 * ib + 2] - cz) * inv;
    float ccx = (verts[3 * ic + 0] - cx) * inv, ccy = (verts[3 * ic + 1] - cy) * inv, ccz = (verts[3 * ic + 2] - cz) * inv;
    float abx = bx - ax, aby = by - ay, abz = bz - az;
    float acx = ccx - ax, acy = ccy - ay, acz = ccz - az;
    float kaba = abx * ax + aby * ay + abz * az;
    float kaca = acx * ax + acy * ay + acz * az;
    float kabb = abx * bx + aby * by + abz * bz;
    float kacb = acx * bx + acy * by + acz * bz;
    float kabc = abx * ccx + aby * ccy + abz * ccz;
    float kacc = acx * ccx + acy * ccy + acz * ccz;
    float lab = abx * abx + aby * aby + abz * abz;
    float lac = acx * acx + acy * acy + acz * acz;
    float lx = abx * acx + aby * acy + abz * acz;
    float aa = ax * ax + ay * ay + az * az;
    float nx = aby * acz - abz * acy;
    float ny = abz * acx - abx * acz;
    float nz = abx * acy - aby * acx;
    float* rec = ws + 4 + 4 * N + (size_t)f * 32;
    ((v4f*)rec)[0] = (v4f){abx, aby, abz, acx};
    ((v4f*)rec)[1] = (v4f){acy, acz, ax, ay};
    ((v4f*)rec)[2] = (v4f){az, kaba, kaca, kabb};
    ((v4f*)rec)[3] = (v4f){kacb, kabc, kacc, lab};
    ((v4f*)rec)[4] = (v4f){lac, lx, aa, nx};
    ((v2f*)rec)[10] = (v2f){ny, nz};
}

// Ericson region logic -> barycentric (u,v) with a single reciprocal.
// Priority cond1 > cond2 > ... > cond6 > interior (applied in reverse).
__device__ __forceinline__ void region_uv(float d1, float d2d, float d3, float d4,
                                          float d5, float d6, float& uo, float& vo) {
    const float vcv = d1 * d4 - d3 * d2d;
    const float vbv = d5 * d2d - d1 * d6;
    const float vav = d3 * d6 - d5 * d4;
    const float e1 = d4 - d3, e2 = d5 - d6;
    float nu = vbv, nv = vcv, den = vav + vbv + vcv;
    bool bc = false;
    bool c6 = (vav <= 0.f) & (e1 >= 0.f) & (e2 >= 0.f);
    nu = c6 ? e1 : nu; nv = c6 ? e1 : nv; den = c6 ? (e1 + e2) : den; bc = c6;
    bool c5 = (vbv <= 0.f) & (d2d >= 0.f) & (d6 <= 0.f);
    nu = c5 ? 0.f : nu; nv = c5 ? d2d : nv; den = c5 ? (d2d - d6) : den; bc = bc && !c5;
    bool c4 = (vcv <= 0.f) & (d1 >= 0.f) & (d3 <= 0.f);
    nu = c4 ? d1 : nu; nv = c4 ? 0.f : nv; den = c4 ? (d1 - d3) : den; bc = bc && !c4;
    bool c3 = (d6 >= 0.f) & (d5 <= d6);
    nu = c3 ? 0.f : nu; nv = c3 ? 1.f : nv; den = c3 ? 1.f : den; bc = bc && !c3;
    bool c2 = (d3 >= 0.f) & (d4 <= d3);
    nu = c2 ? 1.f : nu; nv = c2 ? 0.f : nv; den = c2 ? 1.f : den; bc = bc && !c2;
    bool c1 = (d1 <= 0.f) & (d2d <= 0.f);
    nu = c1 ? 0.f : nu; nv = c1 ? 0.f : nv; den = c1 ? 1.f : den; bc = bc && !c1;
    den = (den == 0.f) ? 1.f : den;
    const float rden = __builtin_amdgcn_rcpf(den);
    float v = nv * rden;
    float u = nu * rden;
    u = bc ? (1.f - v) : u;  // BC edge: u = 1 - t, v = t
    uo = u; vo = v;
}

// ---------------- Pass 4: main WMMA kernel ----------------
__global__ __launch_bounds__(TMAIN) void k_sdf_main(const float* __restrict__ ws,
                                                    float* __restrict__ out,
                                                    int N, int F) {
    const float* qx = ws + 4;
    const float* qy = qx + N;
    const float* qz = qy + N;
    const float* pp = qz + N;
    const float* tb = ws + 4 + 4 * N;   // 128B records

    __shared__ u64 sKey[WAVES][32];

    const u32 tid = threadIdx.x;
    const u32 wave = tid >> 5;
    const u32 lane = tid & 31;
    const bool lo = lane < 16;       // lower half of wave
    const u32 cc = lane & 15;        // WMMA column / half-lane id
    const u32 q0 = blockIdx.x * 32;  // 32 queries per block
    const u32 trisPerWave = (u32)F / WAVES;  // 256
    const u32 tiles = trisPerWave >> 4;      // 16
    const u32 triBase = wave * trisPerWave;

    // --- A operands (16x4 f32): lanes 0-15 hold K0=x,K1=y; lanes 16-31 hold K2=z,K3=pp
    v2f A[2];
#pragma unroll
    for (int j = 0; j < 2; ++j) {
        u32 qm = q0 + 16u * j + cc;
        A[j].x = lo ? qx[qm] : qz[qm];
        A[j].y = lo ? qy[qm] : pp[qm];   // K3 carries |p|^2
    }

    u64 key[2][8];
#pragma unroll
    for (int j = 0; j < 2; ++j)
#pragma unroll
        for (int r = 0; r < 8; ++r) key[j][r] = ~0ULL;

    for (u32 t = 0; t < tiles; ++t) {
        const u32 triC = triBase + (t << 4) + cc;
        const u32 rec = triC << 7;   // 128B record, one address for 5 b128 loads

        const v4f r0 = ld4(tb, rec);        // abx aby abz acx
        const v4f r1 = ld4(tb, rec + 16);   // acy acz ax  ay
        const v4f r2 = ld4(tb, rec + 32);   // az  ab.a ac.a ab.b
        const v4f r3 = ld4(tb, rec + 48);   // ac.b ab.c ac.c |ab|^2
        const v4f r4 = ld4(tb, rec + 64);   // |ac|^2 ab.ac |a|^2 nx

        // B operands (4x16 f32): col = cc; lo lanes rows K0,K1; hi lanes rows K2,K3.
        // Bab/Bac K3 = 0 (pp term cancels); Ba = (-2a, K3=1) so Dr = pp - 2 a.p.
        v2f Bab, Bac, Ba;
        Bab.x = lo ? r0.x : r0.z;            Bab.y = lo ? r0.y : 0.0f;
        Bac.x = lo ? r0.w : r1.y;            Bac.y = lo ? r1.x : 0.0f;
        Ba.x  = -2.0f * (lo ? r1.z : r2.x);  Ba.y  = lo ? (-2.0f * r1.w) : 1.0f;

        const float kaba = r2.y, kaca = r2.z, kabb = r2.w;
        const float kacb = r3.x, kabc = r3.y, kacc = r3.z;
        const float lab = r3.w, lac = r4.x, lx = r4.y, aa = r4.z;

#pragma unroll
        for (int j = 0; j < 2; ++j) {
            v8f z = {};
            v8f Ds = __builtin_amdgcn_wmma_f32_16x16x4_f32(false, A[j], false, Bab,
                                                           (short)0, z, false, false);
            v8f Dt = __builtin_amdgcn_wmma_f32_16x16x4_f32(false, A[j], false, Bac,
                                                           (short)0, z, false, false);
            v8f Dr = __builtin_amdgcn_wmma_f32_16x16x4_f32(false, A[j], false, Ba,
                                                           (short)0, z, false, false);
#pragma unroll
            for (int r = 0; r < 8; ++r) {
                const float s = Ds[r];   // ab . p
                const float tt = Dt[r];  // ac . p
                const float d1 = s - kaba, d2d = tt - kaca;
                const float d3 = s - kabb, d4 = tt - kacb;
                const float d5 = s - kabc, d6 = tt - kacc;
                float u, v;
                region_uv(d1, d2d, d3, d4, d5, d6, u, v);
                // |p - (a + u ab + v ac)|^2 expanded with precomputed norms
                const float pa2 = Dr[r] + aa;   // pp - 2 a.p + |a|^2
                const float d2q = pa2 + u * fmaf(u, lab, -2.f * d1)
                                      + v * fmaf(v, lac, -2.f * d2d)
                                      + 2.f * u * v * lx;
                const u64 k = ((u64)__float_as_uint(d2q) << 32) | (u64)triC;
                key[j][r] = (k < key[j][r]) ? k : key[j][r];
            }
        }
    }

    // ---- cross-lane u64-min within each 16-lane half (D-layout column lanes) ----
#pragma unroll
    for (int j = 0; j < 2; ++j)
#pragma unroll
        for (int r = 0; r < 8; ++r) {
            u64 k = key[j][r];
            for (int m = 1; m <= 8; m <<= 1) {
                u64 o = __shfl_xor(k, m, 32);
                k = (o < k) ? o : k;
            }
            if (cc == (u32)r) sKey[wave][16 * j + r + (lo ? 0 : 8)] = k;
        }
    __syncthreads();

    // ---- cross-wave reduction + epilogue (32 queries, one lane each) ----
    if (tid < 32) {
        u64 best = ~0ULL;
        for (int w = 0; w < WAVES; ++w) {
            u64 k = sKey[w][tid];
            best = (k < best) ? k : best;
        }
        const u32 bi = (u32)(best & 0xffffffffu);
        const u32 rec = bi << 7;
        const v4f r0 = ld4(tb, rec);
        const v4f r1 = ld4(tb, rec + 16);
        const v4f r2 = ld4(tb, rec + 32);
        const v4f r4 = ld4(tb, rec + 64);
        const v2f n2 = ld2(tb, rec + 80);
        const float abx = r0.x, aby = r0.y, abz = r0.z;
        const float acx = r0.w, acy = r1.x, acz = r1.y;
        const float ax = r1.z, ay = r1.w, az = r2.x;
        const u32 q = q0 + tid;
        const float px = qx[q], py = qy[q], pz = qz[q];
        // recompute exact region for the winner
        const float apx = px - ax, apy = py - ay, apz = pz - az;
        const float d1 = abx * apx + aby * apy + abz * apz;
        const float d2d = acx * apx + acy * apy + acz * apz;
        const float bpx = apx - abx, bpy = apy - aby, bpz = apz - abz;
        const float d3 = abx * bpx + aby * bpy + abz * bpz;
        const float d4 = acx * bpx + acy * bpy + acz * bpz;
        const float cpx_ = apx - acx, cpy_ = apy - acy, cpz_ = apz - acz;
        const float d5 = abx * cpx_ + aby * cpy_ + abz * cpz_;
        const float d6 = acx * cpx_ + acy * cpy_ + acz * cpz_;
        float u, v;
        region_uv(d1, d2d, d3, d4, d5, d6, u, v);
        const float cx = ax + u * abx + v * acx;
        const float cy = ay + u * aby + v * acy;
        const float cz = az + u * abz + v * acz;
        const float dx = px - cx, dy = py - cy, dz = pz - cz;
        const float dist = sqrtf(dx * dx + dy * dy + dz * dz);
        const float sg = dx * r4.w + dy * n2.x + dz * n2.y;
        out[q] = (sg < 0.f) ? -dist : dist;
    }
}

extern "C" void kernel_launch(void* const* d_in, const int* in_sizes, int n_in,
                              void* d_out, int out_size, void* d_ws, size_t ws_size,
                              hipStream_t stream) {
    const float* queries = (const float*)d_in[0];
    const float* verts = (const float*)d_in[1];
    const int* faces = (const int*)d_in[2];
    const int N = in_sizes[0] / 3;  // 2048
    const int V = in_sizes[1] / 3;  // 16384
    const int F = in_sizes[2] / 3;  // 8192
    float* ws = (float*)d_ws;
    float* out = (float*)d_out;

    k_bbox<<<1, 256, 0, stream>>>(verts, V, ws);
    k_queries<<<(N + 255) / 256, 256, 0, stream>>>(queries, N, ws);
    k_tris<<<(F + 255) / 256, 256, 0, stream>>>(verts, faces, F, N, ws);
    k_sdf_main<<<N / 32, TMAIN, 0, stream>>>(ws, out, N, F);
}